// PointerNet_77017353552417
// MI455X (gfx1250) — compile-verified
//
#include <hip/hip_runtime.h>
#include <math.h>

typedef float v2f __attribute__((ext_vector_type(2)));
typedef float v8f __attribute__((ext_vector_type(8)));

__device__ __forceinline__ float sigmoidf_(float x) { return 1.0f / (1.0f + expf(-x)); }

// ---------------------------------------------------------------------------
// Generic fp32 WMMA GEMM: out[m][n] = sum_k A[m][k] * W[n][k] + b1[n] + b2[n]
// One wave per 16x16 output tile, K stepped by 4 via v_wmma_f32_16x16x4_f32.
// M-padding handled by clamping the A row (padded D rows are never stored,
// and D row m depends only on A row m, so clamped garbage is harmless).
// Software-pipelined operand loads (depth 2). act==1 applies tanh at store.
// ---------------------------------------------------------------------------
__global__ void gemm_wmma_f32(const float* __restrict__ A, const float* __restrict__ W,
                              const float* __restrict__ b1, const float* __restrict__ b2,
                              float* __restrict__ out, int M, int N, int K, int act)
{
    int gtid  = blockIdx.x * blockDim.x + threadIdx.x;
    int w     = gtid >> 5;
    int lane  = threadIdx.x & 31;
    int ntile = N >> 4;
    int mtile = (M + 15) >> 4;
    if (w >= mtile * ntile) return;
    int mt = w / ntile, nt = w % ntile;

    int mrow = (mt << 4) + (lane & 15);
    if (mrow >= M) mrow = M - 1;            // clamp: no exec-masked loads needed
    int nrow = (nt << 4) + (lane & 15);
    int kh   = (lane >> 4) << 1;            // 0 or 2: which K-pair this half-wave owns
    const float* arow = A + (size_t)mrow * K + kh;
    const float* wrow = W + (size_t)nrow * K + kh;

    v8f c = {0.f, 0.f, 0.f, 0.f, 0.f, 0.f, 0.f, 0.f};
    v2f a0  = *(const v2f*)(arow);
    v2f bb0 = *(const v2f*)(wrow);
#pragma unroll 4
    for (int k = 0; k < K - 4; k += 4) {
        v2f a1  = *(const v2f*)(arow + k + 4);      // prefetch next K-step
        v2f bb1 = *(const v2f*)(wrow + k + 4);
        c = __builtin_amdgcn_wmma_f32_16x16x4_f32(false, a0, false, bb0, (short)0, c, false, false);
        a0 = a1; bb0 = bb1;
    }
    c = __builtin_amdgcn_wmma_f32_16x16x4_f32(false, a0, false, bb0, (short)0, c, false, false);

    int n     = (nt << 4) + (lane & 15);
    int mbase = (mt << 4) + ((lane >> 4) << 3);
    float bias = (b1 ? b1[n] : 0.f) + (b2 ? b2[n] : 0.f);
#pragma unroll
    for (int r = 0; r < 8; ++r) {
        int m = mbase + r;
        if (m < M) {
            float v = c[r] + bias;
            if (act) v = tanhf(v);
            out[(size_t)m * N + n] = v;
        }
    }
}

// ---------------------------------------------------------------------------
// Persistent recurrent encoder scan. 32 WGs x 512 threads (16 waves).
// WG owns 16 hidden units (all 4 gates, all 8 batches). Per step:
//   gates[16b x 64rows] = h[16b x 512] @ Whh_sliceT via WMMA f32 16x16x4,
//   K=512 split over 4 waves per gate tile, partials reduced through LDS,
//   elementwise LSTM update, h published to ping-pong global buffer with
//   device-scope stores, grid-wide monotonic atomic barrier per step.
// ---------------------------------------------------------------------------
#define ENC_WGS 32
__global__ __launch_bounds__(512, 1) void encoder_scan(
    const float* __restrict__ Gx,     // [B*L][2048] precomputed x-gates (+biases)
    const float* __restrict__ Whh,    // [2048][512]
    float* __restrict__ enc_out,      // [B][L][512]
    float* hbuf0, float* hbuf1,       // [8][512] ping-pong h (device scope)
    float* __restrict__ c_final,      // [8][512]
    unsigned int* bar)
{
    const int L = 2048;
    int wg   = blockIdx.x;            // owns units [wg*16, wg*16+16)
    int tid  = threadIdx.x;
    int lane = tid & 31;
    int wave = tid >> 5;              // 0..15
    int gate = wave & 3;              // which gate tile
    int ksub = wave >> 2;             // which K quarter

    __shared__ float hsh[16 * 512];   // A operand [16 batches(8 real)][512], 32KB
    __shared__ float psh[16 * 128];   // per-wave partial C: [wave][n(16)][r(8)]
    __shared__ float csh[16 * 8];     // c state [unit][batch]

    for (int i = tid; i < 16 * 512; i += 512) hsh[i] = 0.f;
    if (tid < 128) csh[tid] = 0.f;

    int u0 = wg << 4;
    int kh = (lane >> 4) << 1;
    const float* wrow = Whh + (size_t)((gate << 9) + u0 + (lane & 15)) * 512 + kh;
    const float* hrow = hsh + ((lane & 15) << 9) + kh;
    int kbase = ksub << 7;            // 128 K per wave
    __syncthreads();

    for (int t = 0; t < L; ++t) {
        if (t > 0) {  // refresh h (batches 0..7) from the buffer written last step
            const float* hr = (t & 1) ? hbuf1 : hbuf0;
            for (int i = tid; i < 8 * 512; i += 512)
                hsh[i] = __hip_atomic_load(hr + i, __ATOMIC_RELAXED, __HIP_MEMORY_SCOPE_AGENT);
        }
        __syncthreads();

        v8f c = {0.f, 0.f, 0.f, 0.f, 0.f, 0.f, 0.f, 0.f};
        v2f a0  = *(const v2f*)(hrow + kbase);
        v2f bb0 = *(const v2f*)(wrow + kbase);
#pragma unroll 4
        for (int k = kbase; k < kbase + 124; k += 4) {
            v2f a1  = *(const v2f*)(hrow + k + 4);
            v2f bb1 = *(const v2f*)(wrow + k + 4);
            c = __builtin_amdgcn_wmma_f32_16x16x4_f32(false, a0, false, bb0, (short)0, c, false, false);
            a0 = a1; bb0 = bb1;
        }
        c = __builtin_amdgcn_wmma_f32_16x16x4_f32(false, a0, false, bb0, (short)0, c, false, false);

        if (lane < 16) {              // lanes 0-15 hold batches 0-7 (r = batch)
            float* p = psh + (wave << 7) + (lane << 3);
#pragma unroll
            for (int r = 0; r < 8; ++r) p[r] = c[r];
        }
        __syncthreads();

        if (tid < 128) {              // unit u = tid>>3, batch b = tid&7
            int u = tid >> 3, b = tid & 7;
            float g[4];
#pragma unroll
            for (int gg = 0; gg < 4; ++gg) {
                float s = Gx[(size_t)(b * L + t) * 2048 + (gg << 9) + u0 + u];
#pragma unroll
                for (int ks = 0; ks < 4; ++ks)
                    s += psh[(((ks << 2) + gg) << 7) + (u << 3) + b];
                g[gg] = s;
            }
            float cp = csh[(u << 3) + b];
            float ct = sigmoidf_(g[1]) * cp + sigmoidf_(g[0]) * tanhf(g[2]);
            float ht = sigmoidf_(g[3]) * tanhf(ct);
            csh[(u << 3) + b] = ct;
            float* hw = ((t + 1) & 1) ? hbuf1 : hbuf0;
            __hip_atomic_store(hw + (b << 9) + u0 + u, ht,
                               __ATOMIC_RELAXED, __HIP_MEMORY_SCOPE_AGENT);
            enc_out[((size_t)b * L + t) * 512 + u0 + u] = ht;
            if (t == L - 1) c_final[(b << 9) + u0 + u] = ct;
        }
        __threadfence();              // make this WG's h stores agent-visible
        __syncthreads();
        if (tid == 0) {
            __hip_atomic_fetch_add(bar, 1u, __ATOMIC_RELEASE, __HIP_MEMORY_SCOPE_AGENT);
            unsigned int target = (unsigned int)(t + 1) * ENC_WGS;
            while (__hip_atomic_load(bar, __ATOMIC_ACQUIRE, __HIP_MEMORY_SCOPE_AGENT) < target)
                __builtin_amdgcn_s_sleep(1);
        }
        __syncthreads();
    }
}

// ---------------------------------------------------------------------------
// Decoder LSTM elementwise: gates = g1 + g2 (biases already folded by GEMMs).
// ---------------------------------------------------------------------------
__global__ void lstm_cell_ew(const float* __restrict__ g1, const float* __restrict__ g2,
                             float* __restrict__ c, float* __restrict__ ht)
{
    int idx = blockIdx.x * blockDim.x + threadIdx.x;
    if (idx >= 8 * 512) return;
    int b = idx >> 9, u = idx & 511;
    const float* r1 = g1 + b * 2048;
    const float* r2 = g2 + b * 2048;
    float gi = r1[u]        + r2[u];
    float gf = r1[512 + u]  + r2[512 + u];
    float gg = r1[1024 + u] + r2[1024 + u];
    float go = r1[1536 + u] + r2[1536 + u];
    float ct = sigmoidf_(gf) * c[idx] + sigmoidf_(gi) * tanhf(gg);
    c[idx]  = ct;
    ht[idx] = sigmoidf_(go) * tanhf(ct);
}

// att[b][l] = sum_h V[h] * tanh(inp[b][h] + ctx[b*L+l][h]) ; one wave per (b,l)
__global__ void attn_scores(const float* __restrict__ inp, const float* __restrict__ ctx,
                            const float* __restrict__ V, float* __restrict__ att)
{
    int gtid = blockIdx.x * blockDim.x + threadIdx.x;
    int w = gtid >> 5, lane = threadIdx.x & 31;
    if (w >= 8 * 2048) return;
    int b = w >> 11;
    const float* crow = ctx + (size_t)w * 512;
    const float* irow = inp + (b << 9);
    float s = 0.f;
    for (int h = lane; h < 512; h += 32)
        s += V[h] * tanhf(irow[h] + crow[h]);
#pragma unroll
    for (int off = 16; off > 0; off >>= 1) s += __shfl_xor(s, off, 32);
    if (lane == 0) att[w] = s;
}

// Per-batch: masked softmax -> alpha (to d_out), argmax -> pointer, mask update,
// hidden_state = sum_l ctx[b][l][:] * alpha[l], concat [hidden_state, ht], x gather.
__global__ __launch_bounds__(256) void softmax_ptr(
    const float* __restrict__ att, float* __restrict__ mask,
    const float* __restrict__ ctx, const float* __restrict__ enc_out,
    const float* __restrict__ ht, float* __restrict__ cc,
    float* __restrict__ xbuf, float* __restrict__ alpha_out,
    float* __restrict__ ptr_out, int step)
{
    const int L = 2048;
    int b = blockIdx.x, tid = threadIdx.x;
    const float* arow = att + b * L;
    float* mrow = mask + b * L;
    float* al = alpha_out + (size_t)(b * 2 + step) * L;
    __shared__ float rv[256];
    __shared__ int   ri[256];
    __shared__ int   sh_idx;

    float m = -3.0e38f;
    for (int l = tid; l < L; l += 256)
        if (mrow[l] != 0.f) m = fmaxf(m, arow[l]);
    rv[tid] = m; __syncthreads();
    for (int s = 128; s > 0; s >>= 1) { if (tid < s) rv[tid] = fmaxf(rv[tid], rv[tid + s]); __syncthreads(); }
    float gmax = rv[0]; __syncthreads();

    float s = 0.f;
    for (int l = tid; l < L; l += 256) {
        float e = (mrow[l] != 0.f) ? expf(arow[l] - gmax) : 0.f;
        al[l] = e; s += e;
    }
    rv[tid] = s; __syncthreads();
    for (int st = 128; st > 0; st >>= 1) { if (tid < st) rv[tid] += rv[tid + st]; __syncthreads(); }
    float inv = 1.f / rv[0]; __syncthreads();

    float bv = -1.f; int bi = 0x7fffffff;
    for (int l = tid; l < L; l += 256) {
        float a = al[l] * inv;
        al[l] = a;
        if (a > bv) { bv = a; bi = l; }
    }
    rv[tid] = bv; ri[tid] = bi; __syncthreads();
    for (int st = 128; st > 0; st >>= 1) {
        if (tid < st) {
            if (rv[tid + st] > rv[tid] || (rv[tid + st] == rv[tid] && ri[tid + st] < ri[tid])) {
                rv[tid] = rv[tid + st]; ri[tid] = ri[tid + st];
            }
        }
        __syncthreads();
    }
    if (tid == 0) {
        sh_idx = ri[0];
        mrow[ri[0]] = 0.f;
        ptr_out[b * 2 + step] = (float)ri[0];
    }
    __syncthreads();
    int idx = sh_idx;

    for (int h = tid; h < 512; h += 256) {
        float acc = 0.f;
        const float* cb = ctx + (size_t)b * L * 512 + h;
        for (int l = 0; l < L; ++l) acc += cb[(size_t)l * 512] * al[l];
        cc[b * 1024 + h]       = acc;
        cc[b * 1024 + 512 + h] = ht[(b << 9) + h];
        xbuf[(b << 9) + h]     = enc_out[((size_t)b * L + idx) * 512 + h];
    }
}

__global__ void init_state(float* mask, unsigned int* bar)
{
    int i = blockIdx.x * blockDim.x + threadIdx.x;
    if (i < 8 * 2048) mask[i] = 1.f;
    if (i == 0) *bar = 0u;
}

// ---------------------------------------------------------------------------
extern "C" void kernel_launch(void* const* d_in, const int* in_sizes, int n_in,
                              void* d_out, int out_size, void* d_ws, size_t ws_size,
                              hipStream_t stream)
{
    const int B = 8, L = 2048, E = 768, H = 512;
    const float* emb   = (const float*)d_in[0];
    const float* dec0  = (const float*)d_in[1];
    const float* W_ih  = (const float*)d_in[2];
    const float* W_hh  = (const float*)d_in[3];
    const float* b_ih  = (const float*)d_in[4];
    const float* b_hh  = (const float*)d_in[5];
    const float* Wd_in = (const float*)d_in[6];
    const float* bd_in = (const float*)d_in[7];
    const float* Wd_hh = (const float*)d_in[8];
    const float* bd_hh = (const float*)d_in[9];
    const float* W_inp = (const float*)d_in[10];
    const float* b_inp = (const float*)d_in[11];
    const float* Wc    = (const float*)d_in[12];
    const float* bc    = (const float*)d_in[13];
    const float* V     = (const float*)d_in[14];
    const float* W_out = (const float*)d_in[15];
    const float* b_out = (const float*)d_in[16];
    float* out = (float*)d_out;

    float* ws = (float*)d_ws;
    size_t off = 0;
    auto alloc = [&](size_t n) { float* p = ws + off; off += n; return p; };
    float* Gx      = alloc((size_t)B * L * 4 * H);   // 33.5M floats
    float* enc_out = alloc((size_t)B * L * H);       // 8.4M
    float* ctx     = alloc((size_t)B * L * H);       // 8.4M
    float* hbuf0   = alloc(B * H);
    float* hbuf1   = alloc(B * H);
    float* dec_c   = alloc(B * H);
    float* g1      = alloc(B * 4 * H);
    float* g2      = alloc(B * 4 * H);
    float* ht      = alloc(B * H);
    float* inp     = alloc(B * H);
    float* att     = alloc(B * L);
    float* cc      = alloc(B * 2 * H);
    float* xbuf    = alloc(B * H);
    float* dech    = alloc(B * H);
    float* dech2   = alloc(B * H);
    float* mask    = alloc(B * L);
    unsigned int* bar = (unsigned int*)(ws + off); off += 1;
    (void)ws_size; (void)in_sizes; (void)n_in; (void)out_size;

    init_state<<<(B * L + 255) / 256, 256, 0, stream>>>(mask, bar);

    auto gemm = [&](const float* A, const float* W, const float* bb1, const float* bb2,
                    float* O, int M, int N, int K, int act) {
        int tiles = ((M + 15) / 16) * (N / 16);
        int blocks = (tiles * 32 + 255) / 256;
        gemm_wmma_f32<<<blocks, 256, 0, stream>>>(A, W, bb1, bb2, O, M, N, K, act);
    };

    // Encoder input projection for all timesteps (+ both biases folded).
    gemm(emb, W_ih, b_ih, b_hh, Gx, B * L, 4 * H, E, 0);
    // Recurrent scan (persistent, WMMA, grid barrier per step).
    encoder_scan<<<ENC_WGS, 512, 0, stream>>>(Gx, W_hh, enc_out, hbuf0, hbuf1, dec_c, bar);
    // Attention context: conv1d(k=1) == per-position linear.
    gemm(enc_out, Wc, bc, nullptr, ctx, B * L, H, H, 0);

    const float* x = dec0;
    const float* hcur = hbuf0;                 // encoder final h
    float* ptr_out = out + (size_t)B * 2 * L;
    for (int step = 0; step < 2; ++step) {
        gemm(x, Wd_in, bd_in, nullptr, g1, B, 4 * H, H, 0);
        gemm(hcur, Wd_hh, bd_hh, nullptr, g2, B, 4 * H, H, 0);
        lstm_cell_ew<<<(B * H + 255) / 256, 256, 0, stream>>>(g1, g2, dec_c, ht);
        gemm(ht, W_inp, b_inp, nullptr, inp, B, H, H, 0);
        attn_scores<<<(B * L * 32 + 255) / 256, 256, 0, stream>>>(inp, ctx, V, att);
        softmax_ptr<<<B, 256, 0, stream>>>(att, mask, ctx, enc_out, ht, cc, xbuf,
                                           out, ptr_out, step);
        float* hnext = (step == 0) ? dech : dech2;
        gemm(cc, W_out, b_out, nullptr, hnext, B, H, 2 * H, 1);
        x = xbuf;
        hcur = hnext;
    }
}